// Optimal_Transport_Layer_83726092468442
// MI455X (gfx1250) — compile-verified
//
#include <hip/hip_runtime.h>

#define NN0   2048
#define NN1   2048
#define NP    2049          // with dustbin row/col
#define PITCH 2064          // padded row pitch (multiple of 16)
#define NPV   513           // float4 count covering 2052 elements (>= NP, padded)
#define NPAD  2052          // NPV*4
#define KD    256           // feature dim
#define TAU_F 1.02f
#define EPS_F 0.8f
#define OT_ITERS 100
#define NORM_F        (-8.317766166719343f)   // -log(4096)
#define LOG_MU_LAST_F (-0.6931471805599453f)  // log(2048) - log(4096)
#define MN_SUM_F      4096.0f
#define THRESH_F      0.2f

typedef float v2f __attribute__((ext_vector_type(2)));
typedef float v8f __attribute__((ext_vector_type(8)));

// ---------------------------------------------------------------------------
// GEMM (16x16 tile per wave, K stepped by 4 via V_WMMA_F32_16X16X4_F32), then
// E[n,m] = exp(dot(A[:,n],B[:,m]) / 16).  Called twice with swapped A/B to
// produce both E and E^T with coalesced stores.
// ---------------------------------------------------------------------------
__global__ void __launch_bounds__(256)
ot_gemm_exp(const float* __restrict__ A, const float* __restrict__ B,
            float* __restrict__ E)
{
    const int tid  = threadIdx.x;
    const int lane = tid & 31;
    const int wave = tid >> 5;
    const int lm   = lane & 15;   // N index within tile (and A's M index)
    const int lh   = lane >> 4;   // high half selector

    const int n0 = (blockIdx.y * 8 + wave) * 16;  // output row block
    const int m0 = blockIdx.x * 16;               // output col block

    v8f acc = {};
    const float* ap = A + n0 + lm;
    const float* bp = B + m0 + lm;

    for (int k = 0; k < KD; k += 4) {
        const int kbase = (k + 2 * lh) * NN0;     // K = 2*lh + t, t = vgpr idx
        v2f a, b;
        a.x = ap[kbase];
        a.y = ap[kbase + NN0];
        b.x = bp[kbase];
        b.y = bp[kbase + NN0];
        acc = __builtin_amdgcn_wmma_f32_16x16x4_f32(
            /*neg_a=*/false, a, /*neg_b=*/false, b,
            /*c_mod=*/(short)0, acc, /*reuse_a=*/false, /*reuse_b=*/false);
    }

    const float sc = 0.0625f;  // 1/sqrt(256)
    float* erow = E + (size_t)(n0 + 8 * lh) * PITCH + m0 + lm;
    #pragma unroll
    for (int t = 0; t < 8; ++t) {
        erow[(size_t)t * PITCH] = __expf(acc[t] * sc);  // row M = 8*lh + t
    }
}

// ---------------------------------------------------------------------------
// Fill dustbin row/col with exp(bin_score); zero pad lanes; init u=v=0, e=1.
// ---------------------------------------------------------------------------
__global__ void ot_init(float* __restrict__ E, float* __restrict__ ET,
                        float* __restrict__ u, float* __restrict__ v,
                        float* __restrict__ eu, float* __restrict__ ev,
                        const float* __restrict__ binscore)
{
    const int i = blockIdx.x * blockDim.x + threadIdx.x;
    if (i >= NP) return;
    const float ea = __expf(binscore[0]);
    E [(size_t)i * PITCH + NN1]   = ea;   // E[i][2048]
    E [(size_t)NN0 * PITCH + i]   = ea;   // E[2048][i]
    ET[(size_t)i * PITCH + NN0]   = ea;
    ET[(size_t)NN1 * PITCH + i]   = ea;
    // zero vec4 pad columns of each row
    #pragma unroll
    for (int c = NP; c < NPAD; ++c) {
        E [(size_t)i * PITCH + c] = 0.f;
        ET[(size_t)i * PITCH + c] = 0.f;
    }
    u[i] = 0.f;  v[i] = 0.f;
    eu[i] = 1.f; ev[i] = 1.f;
    if (i < NPAD - NP) {  // zero vector pad entries 2049..2051
        u[NP + i] = 0.f;  v[NP + i] = 0.f;
        eu[NP + i] = 0.f; ev[NP + i] = 0.f;
    }
}

// ---------------------------------------------------------------------------
// One Sinkhorn half-step.  One wave per row (8 rows per block); ev staged in
// LDS once per block; float4 global loads (b128) for the E row stream.
//   s = sum_c Emat[r][c]*evo[c];  lse = log(s);
//   un = (log_mu[r]-lse)/tau;  us[r] += eps*(un-us[r]);  eus[r] = exp(us[r]).
// ---------------------------------------------------------------------------
__global__ void __launch_bounds__(256)
ot_sinkhorn(const float* __restrict__ Emat, const float* __restrict__ evo,
            float* __restrict__ us, float* __restrict__ eus)
{
    __shared__ float4 sev[NPV];
    const float4* evv = (const float4*)evo;
    for (int i = threadIdx.x; i < NPV; i += 256)
        sev[i] = evv[i];
    __syncthreads();

    const int wave = threadIdx.x >> 5;
    const int lane = threadIdx.x & 31;
    const int row  = blockIdx.x * 8 + wave;
    if (row >= NP) return;

    const float4* rp = (const float4*)(Emat + (size_t)row * PITCH);
    float s = 0.f;
    for (int i = lane; i < NPV; i += 32) {
        const float4 e = rp[i];
        const float4 w = sev[i];
        s += e.x * w.x + e.y * w.y + e.z * w.z + e.w * w.w;
    }
    #pragma unroll
    for (int off = 16; off > 0; off >>= 1)
        s += __shfl_xor(s, off, 32);

    if (lane == 0) {
        const float lse  = __logf(s);
        const float lmu  = (row < NN0) ? NORM_F : LOG_MU_LAST_F;
        const float un   = (lmu - lse) / TAU_F;
        const float uold = us[row];
        const float unew = uold + EPS_F * (un - uold);
        us[row]  = unew;
        eus[row] = __expf(unew);
    }
}

// ---------------------------------------------------------------------------
// expZ[r][c] = E[r][c]*eu[r]*ev[c]*(m+n) written to d_out; per-row max/argmax
// over inner (c < 2048, r < 2048) with first-index tie break.
// ---------------------------------------------------------------------------
__global__ void __launch_bounds__(256)
ot_expz_rowmax(const float* __restrict__ E, const float* __restrict__ eu,
               const float* __restrict__ ev, float* __restrict__ out,
               float* __restrict__ rowmax, int* __restrict__ idx0)
{
    const int r = blockIdx.x;
    const float f = eu[r] * MN_SUM_F;
    const float* rp = E + (size_t)r * PITCH;

    float bv = -1.f; int bi = 0x7fffffff;
    for (int c = threadIdx.x; c < NP; c += 256) {
        const float val = rp[c] * ev[c] * f;
        out[(size_t)r * NP + c] = val;
        if (c < NN1 && val > bv) { bv = val; bi = c; }
    }
    if (r >= NN0) return;  // dustbin row: no argmax

    #pragma unroll
    for (int off = 16; off > 0; off >>= 1) {
        const float ov = __shfl_xor(bv, off, 32);
        const int   oi = __shfl_xor(bi, off, 32);
        if (ov > bv || (ov == bv && oi < bi)) { bv = ov; bi = oi; }
    }
    __shared__ float sv[8];
    __shared__ int   si[8];
    const int lane = threadIdx.x & 31, wave = threadIdx.x >> 5;
    if (lane == 0) { sv[wave] = bv; si[wave] = bi; }
    __syncthreads();
    if (threadIdx.x == 0) {
        #pragma unroll
        for (int w = 1; w < 8; ++w)
            if (sv[w] > bv || (sv[w] == bv && si[w] < bi)) { bv = sv[w]; bi = si[w]; }
        rowmax[r] = bv;
        idx0[r]   = bi;
    }
}

// Column argmax via ET (coalesced): idx1[m] = argmax_n expZ_inner[n][m].
__global__ void __launch_bounds__(256)
ot_colmax(const float* __restrict__ ET, const float* __restrict__ eu,
          const float* __restrict__ ev, int* __restrict__ idx1)
{
    const int m = blockIdx.x;
    const float f = ev[m] * MN_SUM_F;   // constant per column; keeps values = expZ
    const float* rp = ET + (size_t)m * PITCH;

    float bv = -1.f; int bi = 0x7fffffff;
    for (int n = threadIdx.x; n < NN0; n += 256) {
        const float val = rp[n] * eu[n] * f;
        if (val > bv) { bv = val; bi = n; }
    }
    #pragma unroll
    for (int off = 16; off > 0; off >>= 1) {
        const float ov = __shfl_xor(bv, off, 32);
        const int   oi = __shfl_xor(bi, off, 32);
        if (ov > bv || (ov == bv && oi < bi)) { bv = ov; bi = oi; }
    }
    __shared__ float sv[8];
    __shared__ int   si[8];
    const int lane = threadIdx.x & 31, wave = threadIdx.x >> 5;
    if (lane == 0) { sv[wave] = bv; si[wave] = bi; }
    __syncthreads();
    if (threadIdx.x == 0) {
        #pragma unroll
        for (int w = 1; w < 8; ++w)
            if (sv[w] > bv || (sv[w] == bv && si[w] < bi)) { bv = sv[w]; bi = si[w]; }
        idx1[m] = bi;
    }
}

// Mutual check 0: mscores0, valid0, i0.
__global__ void ot_match0(const int* __restrict__ idx0, const int* __restrict__ idx1,
                          const float* __restrict__ rowmax,
                          float* __restrict__ out_i0, float* __restrict__ out_ms0,
                          int* __restrict__ valid0)
{
    const int n = blockIdx.x * blockDim.x + threadIdx.x;
    if (n >= NN0) return;
    const int  j   = idx0[n];
    const bool mut = (idx1[j] == n);
    const float ms = mut ? rowmax[n] : 0.f;     // rowmax already = exp(max Z)
    const bool val = mut && (ms > THRESH_F);
    out_i0[n]  = val ? (float)j : -1.f;
    out_ms0[n] = ms;
    valid0[n]  = val ? 1 : 0;
}

// Mutual check 1: mscores1, i1 (reads mscores0 from d_out).
__global__ void ot_match1(const int* __restrict__ idx0, const int* __restrict__ idx1,
                          const float* __restrict__ ms0, const int* __restrict__ valid0,
                          float* __restrict__ out_i1, float* __restrict__ out_ms1)
{
    const int m = blockIdx.x * blockDim.x + threadIdx.x;
    if (m >= NN1) return;
    const int  i   = idx1[m];
    const bool mut = (idx0[i] == m);
    out_ms1[m] = mut ? ms0[i] : 0.f;
    const bool val = mut && (valid0[i] != 0);
    out_i1[m]  = val ? (float)i : -1.f;
}

// ---------------------------------------------------------------------------
extern "C" void kernel_launch(void* const* d_in, const int* in_sizes, int n_in,
                              void* d_out, int out_size, void* d_ws, size_t ws_size,
                              hipStream_t stream)
{
    (void)in_sizes; (void)n_in; (void)out_size; (void)ws_size;

    const float* md0      = (const float*)d_in[0];  // (1,256,2048)
    const float* md1      = (const float*)d_in[1];  // (1,256,2048)
    const float* binscore = (const float*)d_in[2];  // scalar

    float* out = (float*)d_out;
    float* outZ    = out;                              // 2049*2049
    float* out_i0  = out + (size_t)NP * NP;            // 2048
    float* out_i1  = out_i0 + NN0;                     // 2048
    float* out_ms0 = out_i1 + NN1;                     // 2048
    float* out_ms1 = out_ms0 + NN0;                    // 2048

    // workspace layout (all vectors on PITCH-float strides => 16B aligned)
    char* ws = (char*)d_ws;
    const size_t matBytes = (size_t)PITCH * NP * sizeof(float);
    float* E  = (float*)ws;
    float* ET = (float*)(ws + matBytes);
    float* vecs = (float*)(ws + 2 * matBytes);
    float* u      = vecs + 0 * PITCH;
    float* v      = vecs + 1 * PITCH;
    float* eu     = vecs + 2 * PITCH;
    float* ev     = vecs + 3 * PITCH;
    float* rowmax = vecs + 4 * PITCH;
    int*   idx0   = (int*)(vecs + 5 * PITCH);
    int*   idx1   = (int*)(vecs + 6 * PITCH);
    int*   valid0 = (int*)(vecs + 7 * PITCH);

    // 1) E = exp(scores), ET = exp(scores^T) — two WMMA GEMM passes.
    dim3 gGemm(NN1 / 16, NN0 / (16 * 8));
    ot_gemm_exp<<<gGemm, 256, 0, stream>>>(md0, md1, E);
    ot_gemm_exp<<<gGemm, 256, 0, stream>>>(md1, md0, ET);

    // 2) dustbins + pad zeroing + u/v init.
    ot_init<<<(NP + 255) / 256, 256, 0, stream>>>(E, ET, u, v, eu, ev, binscore);

    // 3) Sinkhorn: 100 iterations, each two alternating half-steps (L2-resident matvec reductions).
    const int skBlocks = (NP + 7) / 8;
    for (int it = 0; it < OT_ITERS; ++it) {
        ot_sinkhorn<<<skBlocks, 256, 0, stream>>>(E,  ev, u, eu);
        ot_sinkhorn<<<skBlocks, 256, 0, stream>>>(ET, eu, v, ev);
    }

    // 4) exp(Z) + row argmax; column argmax via ET.
    ot_expz_rowmax<<<NP, 256, 0, stream>>>(E, eu, ev, outZ, rowmax, idx0);
    ot_colmax<<<NN1, 256, 0, stream>>>(ET, eu, ev, idx1);

    // 5) mutual matching.
    ot_match0<<<(NN0 + 255) / 256, 256, 0, stream>>>(idx0, idx1, rowmax,
                                                     out_i0, out_ms0, valid0);
    ot_match1<<<(NN1 + 255) / 256, 256, 0, stream>>>(idx0, idx1, out_ms0, valid0,
                                                     out_i1, out_ms1);
}